// ProteinTransformer_31628139168411
// MI455X (gfx1250) — compile-verified
//
#include <hip/hip_runtime.h>
#include <math.h>

#define N_NODES 16384
#define KNBR 15
#define CH 128
#define EDGE 24
#define KV_DIM (CH + EDGE)   // 152
#define HEADS 8
#define DHEAD 16
#define DEPTH 3
#define MIX 10
#define RBF_K 16
#define MAXDIST 20.0f

#if defined(__AMDGCN__) && __has_builtin(__builtin_amdgcn_global_load_async_to_lds_b64) && \
    __has_builtin(__builtin_amdgcn_s_wait_asynccnt)
#define HAVE_ASYNC_LDS 1
#else
#define HAVE_ASYNC_LDS 0
#endif

typedef float v2f __attribute__((ext_vector_type(2)));
typedef float v8f __attribute__((ext_vector_type(8)));
typedef int   v2i __attribute__((ext_vector_type(2)));

#if HAVE_ASYNC_LDS
typedef __attribute__((address_space(1))) v2i* gptr_v2i;
typedef __attribute__((address_space(3))) v2i* lptr_v2i;
__device__ __forceinline__ void async_copy_b64(const float* gp, float* lp) {
    __builtin_amdgcn_global_load_async_to_lds_b64((gptr_v2i)gp, (lptr_v2i)lp, 0, 0);
}
#endif

// ---------------------------------------------------------------- geometry
__device__ __forceinline__ void norm3(float& x, float& y, float& z) {
    float r = 1.0f / (sqrtf(x * x + y * y + z * z) + 1e-8f);
    x *= r; y *= r; z *= r;
}

__global__ void frames_kernel(const float* __restrict__ tert, float* __restrict__ frames) {
    int n = blockIdx.x * blockDim.x + threadIdx.x;
    if (n >= N_NODES) return;
    int ib = (n - 1 < 0) ? 0 : n - 1;             // ub = u[max(n-1,0)]
    int iF = (n > N_NODES - 2) ? N_NODES - 2 : n; // uf = u[min(n,N-2)]
    float ubx = tert[(ib + 1) * 12 + 3 + 0] - tert[ib * 12 + 3 + 0];
    float uby = tert[(ib + 1) * 12 + 3 + 1] - tert[ib * 12 + 3 + 1];
    float ubz = tert[(ib + 1) * 12 + 3 + 2] - tert[ib * 12 + 3 + 2];
    norm3(ubx, uby, ubz);
    float ufx = tert[(iF + 1) * 12 + 3 + 0] - tert[iF * 12 + 3 + 0];
    float ufy = tert[(iF + 1) * 12 + 3 + 1] - tert[iF * 12 + 3 + 1];
    float ufz = tert[(iF + 1) * 12 + 3 + 2] - tert[iF * 12 + 3 + 2];
    norm3(ufx, ufy, ufz);
    float bx = ubx - ufx, by = uby - ufy, bz = ubz - ufz;
    norm3(bx, by, bz);
    float nx = uby * ufz - ubz * ufy;
    float ny = ubz * ufx - ubx * ufz;
    float nz = ubx * ufy - uby * ufx;
    norm3(nx, ny, nz);
    float cx = by * nz - bz * ny;
    float cy = bz * nx - bx * nz;
    float cz = bx * ny - by * nx;
    float* F = frames + n * 9;   // F[i*3+j], columns j = {b, n, bxn}
    F[0] = bx; F[1] = nx; F[2] = cx;
    F[3] = by; F[4] = ny; F[5] = cy;
    F[6] = bz; F[7] = nz; F[8] = cz;
}

__global__ void edge_kernel(const float* __restrict__ tert, const float* __restrict__ frames,
                            const int* __restrict__ nbr, float* __restrict__ e) {
    int id = blockIdx.x * blockDim.x + threadIdx.x;
    if (id >= N_NODES * KNBR) return;
    int n = id / KNBR, k = id % KNBR;
    int j = nbr[id];
    float mask = ((n - (k + 1)) >= 0) ? 1.0f : 0.0f;
    float dx = tert[j * 12 + 3 + 0] - tert[n * 12 + 3 + 0];
    float dy = tert[j * 12 + 3 + 1] - tert[n * 12 + 3 + 1];
    float dz = tert[j * 12 + 3 + 2] - tert[n * 12 + 3 + 2];
    float dist = sqrtf(dx * dx + dy * dy + dz * dz);
    float inv = 1.0f / (dist + 1e-8f);
    float hx = dx * inv, hy = dy * inv, hz = dz * inv;
    const float* F = frames + n * 9;
    const float* G = frames + j * 9;
    float* ep = e + id * EDGE;
    const float sigma_inv = (float)RBF_K / MAXDIST;   // 1/1.25
#pragma unroll
    for (int t = 0; t < RBF_K; t++) {
        float mu = (MAXDIST * t) / (float)(RBF_K - 1);
        float z = (dist - mu) * sigma_inv;
        ep[t] = expf(-z * z) * mask;
    }
    ep[16] = (F[0] * hx + F[3] * hy + F[6] * hz) * mask;
    ep[17] = (F[1] * hx + F[4] * hy + F[7] * hz) * mask;
    ep[18] = (F[2] * hx + F[5] * hy + F[8] * hz) * mask;
    float R[3][3];
#pragma unroll
    for (int c = 0; c < 3; c++)
#pragma unroll
        for (int d = 0; d < 3; d++)
            R[c][d] = F[0 * 3 + c] * G[0 * 3 + d] + F[1 * 3 + c] * G[1 * 3 + d] + F[2 * 3 + c] * G[2 * 3 + d];
    float tr = R[0][0] + R[1][1] + R[2][2];
    float q0 = 1.0f + tr;
    float q1 = R[2][1] - R[1][2];
    float q2 = R[0][2] - R[2][0];
    float q3 = R[1][0] - R[0][1];
    float qr = 1.0f / (sqrtf(q0 * q0 + q1 * q1 + q2 * q2 + q3 * q3) + 1e-8f);
    ep[19] = q0 * qr * mask;
    ep[20] = q1 * qr * mask;
    ep[21] = q2 * qr * mask;
    ep[22] = q3 * qr * mask;
    ep[23] = ((float)(j - n) / (float)N_NODES) * mask;
}

// ---------------------------------------------------------------- angle pre-projection
__global__ void pre_kernel(const float* __restrict__ angles, const float* __restrict__ Wp,
                           const float* __restrict__ bp, float* __restrict__ x) {
    int id = blockIdx.x * blockDim.x + threadIdx.x;
    if (id >= N_NODES * CH) return;
    int n = id >> 7, c = id & 127;
    float a0 = 0.f, a1 = 0.f, a2 = 0.f;
    if (n > 0) {
        a0 = angles[(n - 1) * 3 + 0];
        a1 = angles[(n - 1) * 3 + 1];
        a2 = angles[(n - 1) * 3 + 2];
    }
    float f0 = sinf(a0), f1 = sinf(a1), f2 = sinf(a2);
    float f3 = cosf(a0), f4 = cosf(a1), f5 = cosf(a2);
    float acc = bp[c];
    acc += f0 * Wp[0 * CH + c] + f1 * Wp[1 * CH + c] + f2 * Wp[2 * CH + c];
    acc += f3 * Wp[3 * CH + c] + f4 * Wp[4 * CH + c] + f5 * Wp[5 * CH + c];
    x[id] = acc;
}

// ---------------------------------------------------------------- pair-interleave repack
// W (rows x 128, row-major)  ->  P[r/2][col][2]  so one 8B load yields rows (k, k+1) of a column.
__global__ void pack_pairs_kernel(const float* __restrict__ W, float* __restrict__ P, int rows) {
    int i = blockIdx.x * blockDim.x + threadIdx.x;
    if (i >= rows * CH) return;
    int r = i >> 7, c = i & 127;
    P[(r >> 1) * 256 + c * 2 + (r & 1)] = W[i];
}

// ---------------------------------------------------------------- LayerNorm (1 wave32 per row)
__global__ void ln_kernel(const float* __restrict__ x, const float* __restrict__ g,
                          const float* __restrict__ b, float* __restrict__ h) {
    int wave = threadIdx.x >> 5, lane = threadIdx.x & 31;
    int row = blockIdx.x * 8 + wave;
    float4 xv = ((const float4*)(x + row * CH))[lane];
    float s = xv.x + xv.y + xv.z + xv.w;
#pragma unroll
    for (int m = 16; m >= 1; m >>= 1) s += __shfl_xor(s, m);
    float mu = s * (1.0f / 128.0f);
    float d0 = xv.x - mu, d1 = xv.y - mu, d2 = xv.z - mu, d3 = xv.w - mu;
    float v = d0 * d0 + d1 * d1 + d2 * d2 + d3 * d3;
#pragma unroll
    for (int m = 16; m >= 1; m >>= 1) v += __shfl_xor(v, m);
    float r = rsqrtf(v * (1.0f / 128.0f) + 1e-5f);
    float4 gv = ((const float4*)g)[lane];
    float4 bv = ((const float4*)b)[lane];
    float4 o;
    o.x = d0 * r * gv.x + bv.x;
    o.y = d1 * r * gv.y + bv.y;
    o.z = d2 * r * gv.z + bv.z;
    o.w = d3 * r * gv.w + bv.w;
    ((float4*)(h + row * CH))[lane] = o;
}

// ---------------------------------------------------------------- f32 WMMA GEMM, Kd=128
// Y[M x Nd] = X[M x 128] @ W[128 x Nd] + bias (+res, relu). 256 thr, 128x128 tile/block.
// X tile staged via async global->LDS (stride 36, 8B aligned, conflict-free);
// W tile staged pair-interleaved so every B fragment is one ds_load_b64.
__global__ void gemm_f32(const float* __restrict__ X, const float* __restrict__ W,
                         const float* __restrict__ bias, const float* __restrict__ res,
                         float* __restrict__ Y, int Nd, int relu) {
    extern __shared__ float smem[];
    float* Xs  = smem;               // 128 x 36
    float* WsP = smem + 128 * 36;    // 16 kpairs x 128 cols x 2
    const int tid = threadIdx.x;
    const int lane = tid & 31, wave = tid >> 5;
    const int m0 = blockIdx.x * 128;
    const int n0 = blockIdx.y * 128;
    const int mloc = wave * 16;
    const int hh = lane >> 4;        // K half-pair owned by this lane
    const int ml = lane & 15;
    v8f zero = {};
    v8f acc[8];
#pragma unroll
    for (int i = 0; i < 8; i++) acc[i] = zero;

    for (int kc = 0; kc < 128; kc += 32) {
        __syncthreads();
#if HAVE_ASYNC_LDS
        for (int i = tid; i < 128 * 16; i += 256) {     // 8B units
            int r = i >> 4, c2 = i & 15;
            async_copy_b64(X + (size_t)(m0 + r) * CH + kc + c2 * 2,
                           Xs + r * 36 + c2 * 2);
        }
#else
        for (int i = tid; i < 128 * 8; i += 256) {
            int r = i >> 3, c4 = i & 7;
            float4 v = *(const float4*)(X + (size_t)(m0 + r) * CH + kc + c4 * 4);
            float* d = Xs + r * 36 + c4 * 4;
            d[0] = v.x; d[1] = v.y; d[2] = v.z; d[3] = v.w;
        }
#endif
        for (int i = tid; i < 16 * 128; i += 256) {     // kpair x col
            int kp = i >> 7, c = i & 127;
            int gc = n0 + c;
            float w0 = 0.0f, w1 = 0.0f;
            if (gc < Nd) {
                w0 = W[(size_t)(kc + kp * 2) * Nd + gc];
                w1 = W[(size_t)(kc + kp * 2 + 1) * Nd + gc];
            }
            v2f pr; pr.x = w0; pr.y = w1;
            *(v2f*)(WsP + ((size_t)kp * 128 + c) * 2) = pr;
        }
#if HAVE_ASYNC_LDS
        __builtin_amdgcn_s_wait_asynccnt(0);
#endif
        __syncthreads();
#pragma unroll
        for (int kk = 0; kk < 32; kk += 4) {
            v2f a = *(const v2f*)(Xs + (mloc + ml) * 36 + kk + hh * 2);
#pragma unroll
            for (int nt = 0; nt < 8; nt++) {
                v2f bfr = *(const v2f*)(WsP + (((kk >> 1) + hh) * 128 + nt * 16 + ml) * 2);
                acc[nt] = __builtin_amdgcn_wmma_f32_16x16x4_f32(
                    false, a, false, bfr, (short)0, acc[nt], false, false);
            }
        }
    }
    // epilogue: bias (+relu) (+residual)
#pragma unroll
    for (int nt = 0; nt < 8; nt++) {
        int col = n0 + nt * 16 + ml;
        if (col >= Nd) continue;
        float bb = bias[col];
#pragma unroll
        for (int r = 0; r < 8; r++) {
            int row = m0 + mloc + r + (hh << 3);
            float val = acc[nt][r] + bb;
            if (relu) val = fmaxf(val, 0.0f);
            if (res) val += res[(size_t)row * CH + col];
            Y[(size_t)row * Nd + col] = val;
        }
    }
}

// ---------------------------------------------------------------- fused neighbor attention
// Per block: 2 nodes. Async-gather kv=[h[nbr]|e] (32x152) into LDS, WMMA k/v against
// pair-interleaved Wk/Wv (8B global B fragments), then masked softmax + weighted sum.
__global__ void attn_kernel(const float* __restrict__ h, const float* __restrict__ e,
                            const int* __restrict__ nbr, const float* __restrict__ q,
                            const float* __restrict__ WkP, const float* __restrict__ bk,
                            const float* __restrict__ WvP, const float* __restrict__ bv,
                            float* __restrict__ att) {
    extern __shared__ float smem[];
    float* kvs = smem;              // 32 x 156
    float* ks  = kvs + 32 * 156;    // 32 x 132
    float* vs  = ks + 32 * 132;     // 32 x 132
    const int tid = threadIdx.x;
    const int n0 = blockIdx.x * 2;

    // phase 1: gather kv rows into LDS (rows 30/31 zero pad)
#if HAVE_ASYNC_LDS
    for (int i = tid; i < 2 * 156; i += 256) kvs[30 * 156 + i] = 0.0f;   // pad rows
    for (int i = tid; i < 30 * 76; i += 256) {                           // 8B units
        int r = i / 76, c2 = i % 76;
        int nl = r / KNBR, kk2 = r % KNBR;
        int n = n0 + nl;
        const float* gp;
        if (c2 < 64) {
            int j = nbr[n * KNBR + kk2];
            gp = h + (size_t)j * CH + c2 * 2;
        } else {
            gp = e + (size_t)(n * KNBR + kk2) * EDGE + (c2 - 64) * 2;
        }
        async_copy_b64(gp, kvs + r * 156 + c2 * 2);
    }
    __builtin_amdgcn_s_wait_asynccnt(0);
#else
    for (int i = tid; i < 32 * KV_DIM; i += 256) {
        int r = i / KV_DIM, c = i % KV_DIM;
        float val = 0.0f;
        if (r < 2 * KNBR) {
            int nl = r / KNBR, kk2 = r % KNBR;
            int n = n0 + nl;
            if (c < CH) {
                int j = nbr[n * KNBR + kk2];
                val = h[(size_t)j * CH + c];
            } else {
                val = e[(size_t)(n * KNBR + kk2) * EDGE + (c - CH)];
            }
        }
        kvs[r * 156 + c] = val;
    }
#endif
    __syncthreads();

    // phase 2: k = kv@Wk + bk, v = kv@Wv + bv  (waves 0-3 -> k, 4-7 -> v)
    const int lane = tid & 31, wave = tid >> 5;
    const int sel = wave >> 2;
    const int w4 = wave & 3;
    const int mt = w4 & 1;
    const int nbase = (w4 >> 1) * 64;
    const float* Wm = sel ? WvP : WkP;
    const float* bm = sel ? bv : bk;
    float* outs = sel ? vs : ks;
    const int hh = lane >> 4, ml = lane & 15;
    v8f zero = {};
    v8f acc[4];
#pragma unroll
    for (int i = 0; i < 4; i++) acc[i] = zero;
    for (int kk = 0; kk < KV_DIM; kk += 4) {
        v2f a = *(const v2f*)(kvs + (mt * 16 + ml) * 156 + kk + hh * 2);
#pragma unroll
        for (int nt = 0; nt < 4; nt++) {
            v2f bfr = *(const v2f*)(Wm + (size_t)((kk >> 1) + hh) * 256 + (nbase + nt * 16 + ml) * 2);
            acc[nt] = __builtin_amdgcn_wmma_f32_16x16x4_f32(
                false, a, false, bfr, (short)0, acc[nt], false, false);
        }
    }
#pragma unroll
    for (int nt = 0; nt < 4; nt++) {
        int col = nbase + nt * 16 + ml;
        float bb = bm[col];
#pragma unroll
        for (int r = 0; r < 8; r++) {
            int row = mt * 16 + r + (hh << 3);
            outs[row * 132 + col] = acc[nt][r] + bb;
        }
    }
    __syncthreads();

    // phase 3: softmax over 15 neighbors + weighted sum, one thread per (node, head)
    if (tid < 2 * HEADS) {
        int nl = tid >> 3, hd = tid & 7;
        int n = n0 + nl;
        const float* qp = q + (size_t)n * CH + hd * DHEAD;
        float logits[KNBR];
        float wmax = -1e30f;
#pragma unroll
        for (int kk = 0; kk < KNBR; kk++) {
            float l = -1e9f;
            if (n - (kk + 1) >= 0) {
                float s = 0.0f;
                const float* kp = ks + (nl * KNBR + kk) * 132 + hd * DHEAD;
#pragma unroll
                for (int d = 0; d < DHEAD; d++) s += qp[d] * kp[d];
                l = s * 0.25f;   // 1/sqrt(16)
            }
            logits[kk] = l;
            wmax = fmaxf(wmax, l);
        }
        float denom = 0.0f;
#pragma unroll
        for (int kk = 0; kk < KNBR; kk++) {
            logits[kk] = expf(logits[kk] - wmax);
            denom += logits[kk];
        }
        float inv = 1.0f / denom;
        float any = (n >= 1) ? 1.0f : 0.0f;
#pragma unroll
        for (int d = 0; d < DHEAD; d++) {
            float s = 0.0f;
#pragma unroll
            for (int kk = 0; kk < KNBR; kk++)
                s += logits[kk] * vs[(nl * KNBR + kk) * 132 + hd * DHEAD + d];
            att[(size_t)n * CH + hd * DHEAD + d] = s * inv * any;
        }
    }
}

// ---------------------------------------------------------------- mixture heads
__global__ void final_kernel(const float* __restrict__ wt, const float* __restrict__ meanb,
                             const float* __restrict__ concb, const float* __restrict__ facb,
                             const float* __restrict__ angles, float* __restrict__ out) {
    int n = blockIdx.x * blockDim.x + threadIdx.x;
    if (n >= N_NODES) return;
    const float* w = wt + n * MIX;
    float mx = -1e30f;
#pragma unroll
    for (int j = 0; j < MIX; j++) mx = fmaxf(mx, w[j]);
    float ex[MIX];
    float s = 0.0f;
#pragma unroll
    for (int j = 0; j < MIX; j++) { ex[j] = expf(w[j] - mx); s += ex[j]; }
    float inv = 1.0f / s;
    float* o = out + (size_t)n * 100;
#pragma unroll
    for (int j = 0; j < MIX; j++) o[j] = ex[j] * inv;
    float a0 = angles[n * 3 + 0], a1 = angles[n * 3 + 1];
    const float* mb = meanb + n * 30;
    const float* fb = facb + n * 30;
#pragma unroll
    for (int j = 0; j < MIX; j++) o[10 + j] = mb[j];
#pragma unroll
    for (int j = 0; j < MIX; j++) o[20 + j] = mb[10 + j] + fb[j] * a0;
#pragma unroll
    for (int j = 0; j < MIX; j++) o[30 + j] = mb[20 + j] + fb[10 + j] * a0 + fb[20 + j] * a1;
    const float* cb = concb + n * 30;
#pragma unroll
    for (int j = 0; j < 30; j++) o[40 + j] = 0.1f + 1000.0f / (1.0f + expf(-cb[j]));
#pragma unroll
    for (int j = 0; j < 30; j++) o[70 + j] = fb[j];
}

// ---------------------------------------------------------------- host orchestration
extern "C" void kernel_launch(void* const* d_in, const int* in_sizes, int n_in,
                              void* d_out, int out_size, void* d_ws, size_t ws_size,
                              hipStream_t stream) {
    const float* angles = (const float*)d_in[0];
    const float* tert   = (const float*)d_in[1];
    const int*   nbr    = (const int*)d_in[2];
    // d_in[3] nbr_mask: derived analytically from node/neighbor index
    const float* W_pre = (const float*)d_in[4];  const float* b_pre = (const float*)d_in[5];
    const float* W_enc = (const float*)d_in[6];  const float* b_enc = (const float*)d_in[7];
    const float* ln1g  = (const float*)d_in[8];  const float* ln1b  = (const float*)d_in[9];
    const float* ln2g  = (const float*)d_in[10]; const float* ln2b  = (const float*)d_in[11];
    const float* Wq = (const float*)d_in[12]; const float* bq = (const float*)d_in[13];
    const float* Wk = (const float*)d_in[14]; const float* bk = (const float*)d_in[15];
    const float* Wv = (const float*)d_in[16]; const float* bv = (const float*)d_in[17];
    const float* Wo = (const float*)d_in[18]; const float* bo = (const float*)d_in[19];
    const float* W1 = (const float*)d_in[20]; const float* b1 = (const float*)d_in[21];
    const float* W2 = (const float*)d_in[22]; const float* b2 = (const float*)d_in[23];
    const float* W3 = (const float*)d_in[24]; const float* b3 = (const float*)d_in[25];
    const float* Wmean = (const float*)d_in[26]; const float* bmean = (const float*)d_in[27];
    const float* Wconc = (const float*)d_in[28]; const float* bconc = (const float*)d_in[29];
    const float* Wwt   = (const float*)d_in[30]; const float* bwt   = (const float*)d_in[31];
    const float* Wfac  = (const float*)d_in[32]; const float* bfac  = (const float*)d_in[33];

    float* ws = (float*)d_ws;
    float* frames = ws;                                   // N*9
    float* e      = frames + (size_t)N_NODES * 9;         // N*15*24
    float* x      = e + (size_t)N_NODES * KNBR * EDGE;    // N*128
    float* h      = x + (size_t)N_NODES * CH;             // N*128
    float* q      = h + (size_t)N_NODES * CH;             // N*128
    float* attb   = q + (size_t)N_NODES * CH;             // N*128
    float* m1     = attb + (size_t)N_NODES * CH;          // N*128
    float* m2     = m1 + (size_t)N_NODES * CH;            // N*128
    float* meanb  = m2 + (size_t)N_NODES * CH;            // N*30
    float* facb   = meanb + (size_t)N_NODES * 30;         // N*30
    float* concb  = facb + (size_t)N_NODES * 30;          // N*30
    float* wtb    = concb + (size_t)N_NODES * 30;         // N*10
    float* WkP    = wtb + (size_t)N_NODES * MIX;          // 152*128 pair-interleaved
    float* WvP    = WkP + (size_t)KV_DIM * CH;            // 152*128 pair-interleaved

    const size_t gsm  = (128 * 36 + 16 * 128 * 2) * sizeof(float);   // ~34 KB
    const size_t asmb = (32 * 156 + 2 * 32 * 132) * sizeof(float);   // ~54 KB
    dim3 ggrid(N_NODES / 128, 1);

    frames_kernel<<<(N_NODES + 255) / 256, 256, 0, stream>>>(tert, frames);
    edge_kernel<<<(N_NODES * KNBR + 255) / 256, 256, 0, stream>>>(tert, frames, nbr, e);
    pre_kernel<<<(N_NODES * CH + 255) / 256, 256, 0, stream>>>(angles, W_pre, b_pre, h);
    gemm_f32<<<ggrid, 256, gsm, stream>>>(h, W_enc, b_enc, nullptr, x, CH, 0);

    for (int d = 0; d < DEPTH; d++) {
        ln_kernel<<<N_NODES / 8, 256, 0, stream>>>(x, ln1g + d * CH, ln1b + d * CH, h);
        gemm_f32<<<ggrid, 256, gsm, stream>>>(h, Wq + (size_t)d * CH * CH, bq + d * CH, nullptr, q, CH, 0);
        pack_pairs_kernel<<<(KV_DIM * CH + 255) / 256, 256, 0, stream>>>(Wk + (size_t)d * KV_DIM * CH, WkP, KV_DIM);
        pack_pairs_kernel<<<(KV_DIM * CH + 255) / 256, 256, 0, stream>>>(Wv + (size_t)d * KV_DIM * CH, WvP, KV_DIM);
        attn_kernel<<<N_NODES / 2, 256, asmb, stream>>>(h, e, nbr, q, WkP, bk + d * CH, WvP, bv + d * CH, attb);
        gemm_f32<<<ggrid, 256, gsm, stream>>>(attb, Wo + (size_t)d * CH * CH, bo + d * CH, x, x, CH, 0);
        ln_kernel<<<N_NODES / 8, 256, 0, stream>>>(x, ln2g + d * CH, ln2b + d * CH, h);
        gemm_f32<<<ggrid, 256, gsm, stream>>>(h, W1 + (size_t)d * CH * CH, b1 + d * CH, nullptr, m1, CH, 1);
        gemm_f32<<<ggrid, 256, gsm, stream>>>(m1, W2 + (size_t)d * CH * CH, b2 + d * CH, nullptr, m2, CH, 1);
        gemm_f32<<<ggrid, 256, gsm, stream>>>(m2, W3 + (size_t)d * CH * CH, b3 + d * CH, x, x, CH, 0);
    }

    gemm_f32<<<ggrid, 256, gsm, stream>>>(x, Wmean, bmean, nullptr, meanb, 30, 0);
    gemm_f32<<<ggrid, 256, gsm, stream>>>(x, Wfac, bfac, nullptr, facb, 30, 0);
    gemm_f32<<<ggrid, 256, gsm, stream>>>(x, Wconc, bconc, nullptr, concb, 30, 0);
    gemm_f32<<<ggrid, 256, gsm, stream>>>(x, Wwt, bwt, nullptr, wtb, MIX, 0);
    final_kernel<<<(N_NODES + 255) / 256, 256, 0, stream>>>(wtb, meanb, concb, facb, angles, (float*)d_out);
}